// SO2_Contraction_33414845563122
// MI455X (gfx1250) — compile-verified
//
#include <hip/hip_runtime.h>

typedef __attribute__((ext_vector_type(2))) float v2f;
typedef __attribute__((ext_vector_type(8))) float v8f;

#define NB 4      // batch elements per workgroup
#define SW 112    // sbuf width = max NT*16 (ro=3: 7 tiles)

constexpr int p3(int r) { return r == 0 ? 1 : 3 * p3(r - 1); }

// --- per-thread SO2 combo: slice rotated x, embed, inverse-rotate, weight ---
template <int R1, int R2, int K, int RO>
__device__ __forceinline__ void compute_s(const float* xr, float yval, float w,
                                          const float* Ri, float* s) {
  constexpr int J = p3(R1 - K);     // surviving elements of rotated x
  constexpr int STR = p3(K);        // flat stride of trailing-k-zero slice
  constexpr int EMB = p3(R2 - K);   // embed stride (trailing zeros)
  constexpr int NRO = p3(RO);
  constexpr float SC = (K == 0) ? 1.0f
                     : (K == 1) ? 0.57735026918962576f
                     : (K == 2) ? (1.0f / 3.0f)
                                : 0.19245008972987526f;  // 1/sqrt(3^k)
  float u[NRO];
#pragma unroll
  for (int m = 0; m < NRO; m++) u[m] = 0.0f;
#pragma unroll
  for (int j = 0; j < J; j++) u[j * EMB] = xr[j * STR] * yval * SC;
  // rotate_all(rot_inv, u, RO): RO passes of per-axis rotation
#pragma unroll
  for (int p = 0; p < RO; p++) {
    if constexpr (RO > 0) {
      constexpr int REST = NRO / 3;
      float tmp[NRO];
#pragma unroll
      for (int z = 0; z < 3; z++)
#pragma unroll
        for (int t = 0; t < REST; t++)
          tmp[z * REST + t] = Ri[z * 3 + 0] * u[t * 3 + 0] +
                              Ri[z * 3 + 1] * u[t * 3 + 1] +
                              Ri[z * 3 + 2] * u[t * 3 + 2];
#pragma unroll
      for (int m = 0; m < NRO; m++) u[m] = tmp[m];
    }
  }
#pragma unroll
  for (int m = 0; m < NRO; m++) s[m] = u[m] * w;
}

// per-combo channel weight: cw[b, i*64+ch] = cutoff * <hidden, Wm1 row>
__device__ __forceinline__ float combo_w(const float* __restrict__ Wm1,
                                         const float (*hid)[64], int bl, int ch,
                                         float cut, int i) {
  const float* wrow = Wm1 + (size_t)(i * 64 + ch) * 64;
  float a = 0.0f;
#pragma unroll
  for (int kk = 0; kk < 64; kk++) a += hid[bl][kk] * wrow[kk];
  return a * cut;
}

// --- WMMA accumulation: D[16h x 16(b,m)] += Wd[16h x 4c] * sbuf[4c x 16(b,m)]
// A-fragments hoisted to registers once per combo (identical across n-tiles).
template <int NT>
__device__ __forceinline__ void wmma_accum(v8f* acc, float (*sb)[SW],
                                           const float* __restrict__ Wd, int fan,
                                           int pos, int lane, int mt, int ngrp) {
  const int lh = lane >> 4;      // lane half selects K pair (ISA A/B layout)
  const int ll = lane & 15;
  const int arow = mt * 16 + ll; // A: M = lane%16
  if (ngrp < NT) {               // wave-uniform: this wave has tiles to do
    v2f a[16];
    const v2f* wd2 = (const v2f*)(Wd + (size_t)arow * fan + pos * 64);
#pragma unroll
    for (int ks = 0; ks < 16; ks++) a[ks] = wd2[ks * 2 + lh];  // K=4ks+2lh(+1)
#pragma unroll
    for (int ti = 0; ti < (NT + 1) / 2; ti++) {
      const int n = ti * 2 + ngrp;     // wave-uniform
      if (n < NT) {
        const int col = n * 16 + ll;   // B/D: N = lane%16
#pragma unroll
        for (int ks = 0; ks < 16; ks++) {
          const int kb = ks * 4 + 2 * lh;
          v2f b;
          b.x = sb[kb][col];
          b.y = sb[kb + 1][col];
          acc[ti] = __builtin_amdgcn_wmma_f32_16x16x4_f32(
              false, a[ks], false, b, (short)0, acc[ti], false, false);
        }
      }
    }
  }
}

template <int RO, int NT>
__device__ __forceinline__ void write_out(const v8f* acc, float* __restrict__ out,
                                          size_t off, int bgbase, int lane, int mt,
                                          int ngrp) {
  constexpr int NRO = p3(RO);
  const int lh = lane >> 4, ll = lane & 15;
#pragma unroll
  for (int ti = 0; ti < (NT + 1) / 2; ti++) {
    const int n = ti * 2 + ngrp;
    if (n < NT) {
      const int col = n * 16 + ll;
      if (col < NB * NRO) {
        const int blc = col / NRO;
        const int m = col % NRO;
        const size_t base = off + ((size_t)(bgbase + blc) * 64) * NRO + (size_t)m;
#pragma unroll
        for (int v = 0; v < 8; v++) {
          const int h = mt * 16 + v + 8 * lh;  // D: M = vgpr + 8*(lane/16)
          out[base + (size_t)h * NRO] = acc[ti][v];
        }
      }
    }
  }
}

__global__ __launch_bounds__(256) void so2_fused(
    const float* __restrict__ x0, const float* __restrict__ y0,
    const float* __restrict__ x1, const float* __restrict__ y1,
    const float* __restrict__ x2, const float* __restrict__ y2,
    const float* __restrict__ x3, const float* __restrict__ y3,
    const float* __restrict__ radial, const float* __restrict__ cutoff,
    const float* __restrict__ rotm, const float* __restrict__ rotmi,
    const float* __restrict__ Wm0, const float* __restrict__ Wm1,
    const float* __restrict__ Wd0, const float* __restrict__ Wd1,
    const float* __restrict__ Wd2, const float* __restrict__ Wd3,
    float* __restrict__ out) {
  __shared__ float hid[NB][64];
  __shared__ float sbuf[64][SW];

  const int tid = threadIdx.x;
  const int bl = tid >> 6;          // local batch 0..3
  const int ch = tid & 63;          // channel
  const int bg = blockIdx.x * NB + bl;
  const int lane = tid & 31;
  const int wv = tid >> 5;
  const int mt = wv & 3;            // h-tile
  const int ngrp = wv >> 2;         // n-tile parity group

  // hidden = silu(radial @ Wm0^T)
  {
    float a = 0.0f;
#pragma unroll
    for (int q = 0; q < 8; q++) a += radial[(size_t)bg * 8 + q] * Wm0[ch * 8 + q];
    hid[bl][ch] = a / (1.0f + __expf(-a));
  }
  __syncthreads();

  const float cut = cutoff[bg];

  // rotation matrices (wave-uniform loads, L2/L0 cached)
  float R[9], Ri[9];
#pragma unroll
  for (int q = 0; q < 9; q++) {
    R[q] = rotm[(size_t)bg * 9 + q];
    Ri[q] = rotmi[(size_t)bg * 9 + q];
  }

  // load x and rotate all axes by R (rotate_all(rot_mat, x_r, r))
  float xr0a[1];
  xr0a[0] = x0[(size_t)bg * 64 + ch];
  float xr1[3], xr2[9], xr3[27];
  {
    const size_t o1 = ((size_t)bg * 64 + ch) * 3;
#pragma unroll
    for (int q = 0; q < 3; q++) xr1[q] = x1[o1 + q];
    float t1[3];
#pragma unroll
    for (int z = 0; z < 3; z++)
      t1[z] = R[z * 3] * xr1[0] + R[z * 3 + 1] * xr1[1] + R[z * 3 + 2] * xr1[2];
#pragma unroll
    for (int q = 0; q < 3; q++) xr1[q] = t1[q];
  }
  {
    const size_t o2 = ((size_t)bg * 64 + ch) * 9;
#pragma unroll
    for (int q = 0; q < 9; q++) xr2[q] = x2[o2 + q];
#pragma unroll
    for (int p = 0; p < 2; p++) {
      float t2[9];
#pragma unroll
      for (int z = 0; z < 3; z++)
#pragma unroll
        for (int t = 0; t < 3; t++)
          t2[z * 3 + t] = R[z * 3] * xr2[t * 3] + R[z * 3 + 1] * xr2[t * 3 + 1] +
                          R[z * 3 + 2] * xr2[t * 3 + 2];
#pragma unroll
      for (int q = 0; q < 9; q++) xr2[q] = t2[q];
    }
  }
  {
    const size_t o3 = ((size_t)bg * 64 + ch) * 27;
#pragma unroll
    for (int q = 0; q < 27; q++) xr3[q] = x3[o3 + q];
#pragma unroll
    for (int p = 0; p < 3; p++) {
      float t3[27];
#pragma unroll
      for (int z = 0; z < 3; z++)
#pragma unroll
        for (int t = 0; t < 9; t++)
          t3[z * 9 + t] = R[z * 3] * xr3[t * 3] + R[z * 3 + 1] * xr3[t * 3 + 1] +
                          R[z * 3 + 2] * xr3[t * 3 + 2];
#pragma unroll
      for (int q = 0; q < 27; q++) xr3[q] = t3[q];
    }
  }

  // y leading values y_r[b,c,0,...,0]
  float yv[4];
  yv[0] = y0[(size_t)bg * 64 + ch];
  yv[1] = y1[((size_t)bg * 64 + ch) * 3];
  yv[2] = y2[((size_t)bg * 64 + ch) * 9];
  yv[3] = y3[((size_t)bg * 64 + ch) * 27];

#define COMBO(I, R1, R2, K, RO, POS, WD, FAN, NT, XR)                       \
  do {                                                                      \
    const float w_ = combo_w(Wm1, hid, bl, ch, cut, I);                     \
    float s_[p3(RO)];                                                       \
    compute_s<R1, R2, K, RO>(XR, yv[R2], w_, Ri, s_);                       \
    __syncthreads(); /* prior WMMA reads of sbuf done */                    \
    _Pragma("unroll")                                                       \
    for (int m = 0; m < p3(RO); m++)                                        \
      sbuf[ch][bl * p3(RO) + m] = s_[m];                                    \
    __syncthreads(); /* sbuf visible */                                     \
    wmma_accum<NT>(acc, sbuf, WD, FAN, POS, lane, mt, ngrp);                \
  } while (0)

  // ---------- RO = 0 (combos i=0,6,14,22), NT=1 ----------
  {
    v8f acc[1];
#pragma unroll
    for (int v = 0; v < 8; v++) acc[0][v] = 0.0f;
    __syncthreads();
    for (int idx = tid; idx < 64 * 12; idx += 256)  // zero pad cols [4,16)
      sbuf[idx / 12][4 + idx % 12] = 0.0f;
    COMBO(0, 0, 0, 0, 0, 0, Wd0, 256, 1, xr0a);
    COMBO(6, 1, 1, 1, 0, 1, Wd0, 256, 1, xr1);
    COMBO(14, 2, 2, 2, 0, 2, Wd0, 256, 1, xr2);
    COMBO(22, 3, 3, 3, 0, 3, Wd0, 256, 1, xr3);
    write_out<0, 1>(acc, out, (size_t)0, blockIdx.x * NB, lane, mt, ngrp);
  }

  // ---------- RO = 1 (i=1,4,8,12,16,20), NT=1 ----------
  {
    v8f acc[1];
#pragma unroll
    for (int v = 0; v < 8; v++) acc[0][v] = 0.0f;
    __syncthreads();
    for (int idx = tid; idx < 64 * 4; idx += 256)  // zero pad cols [12,16)
      sbuf[idx / 4][12 + idx % 4] = 0.0f;
    COMBO(1, 0, 1, 0, 1, 0, Wd1, 384, 1, xr0a);
    COMBO(4, 1, 0, 0, 1, 1, Wd1, 384, 1, xr1);
    COMBO(8, 1, 2, 1, 1, 2, Wd1, 384, 1, xr1);
    COMBO(12, 2, 1, 1, 1, 3, Wd1, 384, 1, xr2);
    COMBO(16, 2, 3, 2, 1, 4, Wd1, 384, 1, xr2);
    COMBO(20, 3, 2, 2, 1, 5, Wd1, 384, 1, xr3);
    write_out<1, 1>(acc, out, (size_t)8192 * 64, blockIdx.x * NB, lane, mt, ngrp);
  }

  // ---------- RO = 2 (i=2,5,9,10,13,18,21), NT=3 ----------
  {
    v8f acc[2];
#pragma unroll
    for (int ti = 0; ti < 2; ti++)
#pragma unroll
      for (int v = 0; v < 8; v++) acc[ti][v] = 0.0f;
    __syncthreads();
    for (int idx = tid; idx < 64 * 12; idx += 256)  // zero pad cols [36,48)
      sbuf[idx / 12][36 + idx % 12] = 0.0f;
    COMBO(2, 0, 2, 0, 2, 0, Wd2, 448, 3, xr0a);
    COMBO(5, 1, 1, 0, 2, 1, Wd2, 448, 3, xr1);
    COMBO(9, 1, 3, 1, 2, 2, Wd2, 448, 3, xr1);
    COMBO(10, 2, 0, 0, 2, 3, Wd2, 448, 3, xr2);
    COMBO(13, 2, 2, 1, 2, 4, Wd2, 448, 3, xr2);
    COMBO(18, 3, 1, 1, 2, 5, Wd2, 448, 3, xr3);
    COMBO(21, 3, 3, 2, 2, 6, Wd2, 448, 3, xr3);
    write_out<2, 3>(acc, out, (size_t)8192 * 64 * 4, blockIdx.x * NB, lane, mt, ngrp);
  }

  // ---------- RO = 3 (i=3,7,11,15,17,19), NT=7 ----------
  {
    v8f acc[4];
#pragma unroll
    for (int ti = 0; ti < 4; ti++)
#pragma unroll
      for (int v = 0; v < 8; v++) acc[ti][v] = 0.0f;
    __syncthreads();
    for (int idx = tid; idx < 64 * 4; idx += 256)  // zero pad cols [108,112)
      sbuf[idx / 4][108 + idx % 4] = 0.0f;
    COMBO(3, 0, 3, 0, 3, 0, Wd3, 384, 7, xr0a);
    COMBO(7, 1, 2, 0, 3, 1, Wd3, 384, 7, xr1);
    COMBO(11, 2, 1, 0, 3, 2, Wd3, 384, 7, xr2);
    COMBO(15, 2, 3, 1, 3, 3, Wd3, 384, 7, xr2);
    COMBO(17, 3, 0, 0, 3, 4, Wd3, 384, 7, xr3);
    COMBO(19, 3, 2, 1, 3, 5, Wd3, 384, 7, xr3);
    write_out<3, 7>(acc, out, (size_t)8192 * 64 * 13, blockIdx.x * NB, lane, mt, ngrp);
  }
#undef COMBO
}

extern "C" void kernel_launch(void* const* d_in, const int* in_sizes, int n_in,
                              void* d_out, int out_size, void* d_ws, size_t ws_size,
                              hipStream_t stream) {
  // setup_inputs() dict order: x0,y0,x1,y1,x2,y2,x3,y3,radial,cutoff,
  //                            rot_mat,rot_mat_inv,W_mlp0,W_mlp1,Wd0..Wd3
  const float* x0 = (const float*)d_in[0];
  const float* y0 = (const float*)d_in[1];
  const float* x1 = (const float*)d_in[2];
  const float* y1 = (const float*)d_in[3];
  const float* x2 = (const float*)d_in[4];
  const float* y2 = (const float*)d_in[5];
  const float* x3 = (const float*)d_in[6];
  const float* y3 = (const float*)d_in[7];
  const float* radial = (const float*)d_in[8];
  const float* cutoff = (const float*)d_in[9];
  const float* rotm = (const float*)d_in[10];
  const float* rotmi = (const float*)d_in[11];
  const float* Wm0 = (const float*)d_in[12];
  const float* Wm1 = (const float*)d_in[13];
  const float* Wd0 = (const float*)d_in[14];
  const float* Wd1 = (const float*)d_in[15];
  const float* Wd2 = (const float*)d_in[16];
  const float* Wd3 = (const float*)d_in[17];
  so2_fused<<<8192 / NB, 256, 0, stream>>>(x0, y0, x1, y1, x2, y2, x3, y3, radial,
                                           cutoff, rotm, rotmi, Wm0, Wm1, Wd0, Wd1,
                                           Wd2, Wd3, (float*)d_out);
}